// EngryEncoder_30889404793022
// MI455X (gfx1250) — compile-verified
//
#include <hip/hip_runtime.h>

#define N_SIG      1048576
#define PAD        512
#define CHUNK      256
#define CHUNKS_PAD 4112          // 257 waves * 16 chunks (>= 4100 real chunks)
#define WAVES_PER_BATCH 257
#define BATCH      16
#define FRAMES     4096

typedef float v2f __attribute__((ext_vector_type(2)));
typedef float v8f __attribute__((ext_vector_type(8)));

// Map padded position p in [0, CHUNKS_PAD*256) to a valid signal index with
// reflect ('reflect' mode: no edge repeat):  i = (N-1) - |(N-1) - |p - PAD||
__device__ __forceinline__ int reflect_idx(int p) {
    int i = p - PAD;
    i = (i < 0) ? -i : i;                    // left reflect
    int d = (N_SIG - 1) - i;
    d = (d < 0) ? -d : d;                    // right reflect
    return (N_SIG - 1) - d;
}

__device__ __forceinline__ v8f wmma_sq_acc(v2f a, v8f c) {
    // D = A * B + C with B == A (symmetric layout): diagonal accumulates sum(a^2)
    return __builtin_amdgcn_wmma_f32_16x16x4_f32(
        /*neg_a=*/false, a, /*neg_b=*/false, a,
        /*c_mod=*/(short)0, c, /*reuse_a=*/false, /*reuse_b=*/false);
}

// Stage 1: per-wave, compute sum-of-squares of 16 hop chunks (256 samples each)
// on the diagonal of a 16x16 f32 WMMA accumulator.
__global__ void __launch_bounds__(256)
chunk_sumsq_kernel(const float* __restrict__ sig, float* __restrict__ S) {
    const int lane  = threadIdx.x & 31;
    const int gw    = blockIdx.x * (blockDim.x >> 5) + (threadIdx.x >> 5);
    const int batch = gw / WAVES_PER_BATCH;
    const int wib   = gw - batch * WAVES_PER_BATCH;   // wave-in-batch: 0..256
    const int chunkbase = wib * 16;

    const int m    = lane & 15;   // chunk row this lane feeds (A row = B col)
    const int half = lane >> 4;   // K-half: lanes 0-15 hold K0,K1; 16-31 hold K2,K3
    const float* sb = sig + (size_t)batch * N_SIG;

    v8f c0 = {0.f,0.f,0.f,0.f,0.f,0.f,0.f,0.f};
    v8f c1 = c0, c2 = c0, c3 = c0;

    if (wib >= 1 && wib <= 255) {
        // Interior: every padded position maps to p-512 directly; contiguous loads.
        const float* lp = sb - PAD + (chunkbase + m) * CHUNK + 2 * half;
        #pragma unroll 4
        for (int kb = 0; kb < CHUNK; kb += 16) {
            v2f a0 = *(const v2f*)(lp + kb);
            v2f a1 = *(const v2f*)(lp + kb + 4);
            v2f a2 = *(const v2f*)(lp + kb + 8);
            v2f a3 = *(const v2f*)(lp + kb + 12);
            c0 = wmma_sq_acc(a0, c0);
            c1 = wmma_sq_acc(a1, c1);
            c2 = wmma_sq_acc(a2, c2);
            c3 = wmma_sq_acc(a3, c3);
        }
    } else {
        // Edge waves: per-element reflect addressing (wave-uniform branch, EXEC full).
        const int p0 = (chunkbase + m) * CHUNK + 2 * half;
        #pragma unroll 4
        for (int kb = 0; kb < CHUNK; kb += 4) {
            v2f a;
            a.x = sb[reflect_idx(p0 + kb)];
            a.y = sb[reflect_idx(p0 + kb + 1)];
            c0 = wmma_sq_acc(a, c0);
        }
    }

    v8f c = c0 + c1 + c2 + c3;

    // Extract diagonal: (m,m) is VGPR m / lane m for m<8, VGPR m-8 / lane m+16 for m>=8.
    float diag = c[0];                                  // lanes 0 and 24
    diag = (lane == 1 || lane == 25) ? c[1] : diag;
    diag = (lane == 2 || lane == 26) ? c[2] : diag;
    diag = (lane == 3 || lane == 27) ? c[3] : diag;
    diag = (lane == 4 || lane == 28) ? c[4] : diag;
    diag = (lane == 5 || lane == 29) ? c[5] : diag;
    diag = (lane == 6 || lane == 30) ? c[6] : diag;
    diag = (lane == 7 || lane == 31) ? c[7] : diag;

    if (lane < 8) {
        S[batch * CHUNKS_PAD + chunkbase + lane] = diag;           // chunks m=0..7
    } else if (lane >= 24) {
        S[batch * CHUNKS_PAD + chunkbase + (lane - 16)] = diag;    // chunks m=8..15
    }
}

// Stage 2: energy[b][f] = sqrt((S[f]+S[f+1]+S[f+2]+S[f+3]) / 1024)
__global__ void __launch_bounds__(256)
energy_kernel(const float* __restrict__ S, float* __restrict__ out) {
    const int t = blockIdx.x * blockDim.x + threadIdx.x;   // 0 .. 65535
    const int b = t >> 12;
    const int f = t & (FRAMES - 1);
    const float* Sb = S + b * CHUNKS_PAD + f;
    float s = Sb[0] + Sb[1] + Sb[2] + Sb[3];
    out[t] = sqrtf(s * (1.0f / 1024.0f));
}

extern "C" void kernel_launch(void* const* d_in, const int* in_sizes, int n_in,
                              void* d_out, int out_size, void* d_ws, size_t ws_size,
                              hipStream_t stream) {
    const float* sig = (const float*)d_in[0];
    float* out = (float*)d_out;
    float* S   = (float*)d_ws;   // 16 * 4112 floats = 263168 bytes

    // 514 blocks * 8 waves = 4112 waves = 16 batches * 257 waves/batch
    chunk_sumsq_kernel<<<514, 256, 0, stream>>>(sig, S);
    energy_kernel<<<(BATCH * FRAMES) / 256, 256, 0, stream>>>(S, out);
}